// WineQualitySNN_2448131359359
// MI455X (gfx1250) — compile-verified
//
#include <hip/hip_runtime.h>
#include <stdint.h>

typedef __attribute__((ext_vector_type(2))) float v2f;
typedef __attribute__((ext_vector_type(8))) float v8f;

#define T_STEPS 25
#define D_IN    57
#define H1      28
#define H2      8
#define O_OUT   3
#define BETA    0.9f
#define THRESH  1.0f

#define ROWS_PER_BLOCK  64
#define WAVES_PER_BLOCK 4
#define THREADS         (WAVES_PER_BLOCK * 32)
#define XTILE_FLOATS    (ROWS_PER_BLOCK * D_IN)   // 3648 floats = 14592 B (contiguous span of x)
#define XTILE_VEC16     (XTILE_FLOATS * 4 / 16)   // 912 16-byte packets
#define NCH1 15   // ceil(57/4)
#define NCH2 7    // 28/4
#define NCH3 2    // 8/4

// D = A(16x4 f32) * B(4x16 f32) + C(16x16 f32)
__device__ __forceinline__ v8f wmma_f32_16x16x4(v2f a, v2f b, v8f c) {
  return __builtin_amdgcn_wmma_f32_16x16x4_f32(false, a, false, b, (short)0, c,
                                               false, false);
}

// Async HBM -> LDS DMA, 16 bytes per lane, tracked by ASYNCcnt.
__device__ __forceinline__ void async_copy_b128(uint32_t lds_byte_addr, const void* gaddr) {
  asm volatile("global_load_async_to_lds_b128 %0, %1, off"
               :: "v"(lds_byte_addr), "v"(gaddr)
               : "memory");
}

__device__ __forceinline__ void wait_asynccnt0() {
  asm volatile("s_wait_asynccnt 0x0" ::: "memory");
}

__global__ __launch_bounds__(THREADS) void snn_lif_wmma_kernel(
    const float* __restrict__ x,
    const float* __restrict__ w1, const float* __restrict__ b1,
    const float* __restrict__ w2, const float* __restrict__ b2,
    const float* __restrict__ w3, const float* __restrict__ b3,
    float* __restrict__ out, int batch)
{
  __shared__ __align__(16) float xtile[2][XTILE_FLOATS];          // double-buffered x tile
  __shared__ __align__(16) float s1t[WAVES_PER_BLOCK][16 * H1];   // per-wave spike tile L1
  __shared__ __align__(16) float s2t[WAVES_PER_BLOCK][16 * 9];    // per-wave spike tile L2 (padded)

  const int tid  = threadIdx.x;
  const int wave = tid >> 5;
  const int lane = tid & 31;
  const int col  = lane & 15;   // N / column index of this lane in C layout
  const int hi   = lane >> 4;   // 0: K+0/1, rows 0..7 ; 1: K+2/3, rows 8..15
  const int rowBase = blockIdx.x * ROWS_PER_BLOCK;
  const int waveRow = rowBase + wave * 16;

  const size_t TB = (size_t)T_STEPS * (size_t)batch;
  float* outM3 = out;                       // mem_rec3 (T,B,3)
  float* outS3 = out + TB * O_OUT;          // spk_rec3 (T,B,3)
  float* outS1 = out + TB * O_OUT * 2;      // spk_rec1 (T,B,28)
  float* outS2 = outS1 + TB * H1;           // spk_rec2 (T,B,8)

  // ---- Preload B-matrix fragments (w^T, zero-padded) into registers, reused all 25 steps ----
  // B frag layout (4x16 f32): lane col = N; v.x = B[kb][n], v.y = B[kb+1][n], kb = 4c + 2*hi.
  v2f bw1[NCH1][2];
#pragma unroll
  for (int c = 0; c < NCH1; ++c) {
#pragma unroll
    for (int nt = 0; nt < 2; ++nt) {
      int n = nt * 16 + col;
      int k = c * 4 + hi * 2;
      v2f b;
      b.x = (n < H1 && k     < D_IN) ? w1[n * D_IN + k]     : 0.f;
      b.y = (n < H1 && k + 1 < D_IN) ? w1[n * D_IN + k + 1] : 0.f;
      bw1[c][nt] = b;
    }
  }
  v2f bw2[NCH2];
#pragma unroll
  for (int c = 0; c < NCH2; ++c) {
    int n = col, k = c * 4 + hi * 2;
    v2f b;
    b.x = (n < H2 && k     < H1) ? w2[n * H1 + k]     : 0.f;
    b.y = (n < H2 && k + 1 < H1) ? w2[n * H1 + k + 1] : 0.f;
    bw2[c] = b;
  }
  v2f bw3[NCH3];
#pragma unroll
  for (int c = 0; c < NCH3; ++c) {
    int n = col, k = c * 4 + hi * 2;
    v2f b;
    b.x = (n < O_OUT && k     < H2) ? w3[n * H2 + k]     : 0.f;
    b.y = (n < O_OUT && k + 1 < H2) ? w3[n * H2 + k + 1] : 0.f;
    bw3[c] = b;
  }
  float bias1[2];
  bias1[0] = (col      < H1) ? b1[col]      : 0.f;
  bias1[1] = (16 + col < H1) ? b1[16 + col] : 0.f;
  float bias2 = (col < H2)    ? b2[col] : 0.f;
  float bias3 = (col < O_OUT) ? b3[col] : 0.f;

  // Membrane state in WMMA C layout: element (row = vr + 8*hi, col) per lane.
  v8f mem1[2] = {{}, {}};
  v8f mem2 = {};
  v8f mem3 = {};

  uint32_t ldsX[2];
  ldsX[0] = (uint32_t)(uintptr_t)(&xtile[0][0]);  // flat LDS addr low 32 bits = LDS offset
  ldsX[1] = (uint32_t)(uintptr_t)(&xtile[1][0]);

  // Kick off DMA for t = 0.
  {
    const char* src = (const char*)(x + ((size_t)rowBase) * D_IN);
    for (int i = tid; i < XTILE_VEC16; i += THREADS)
      async_copy_b128(ldsX[0] + (uint32_t)i * 16u, src + (size_t)i * 16);
  }

  const int r = col;  // A-matrix row handled by this lane

  for (int t = 0; t < T_STEPS; ++t) {
    const int buf = t & 1;
    wait_asynccnt0();
    __syncthreads();  // whole x tile for step t is now visible to all waves

    // Prefetch step t+1 into the other buffer (everyone is done reading it).
    if (t + 1 < T_STEPS) {
      const char* src = (const char*)(x + ((size_t)(t + 1) * batch + rowBase) * D_IN);
      for (int i = tid; i < XTILE_VEC16; i += THREADS)
        async_copy_b128(ldsX[buf ^ 1] + (uint32_t)i * 16u, src + (size_t)i * 16);
    }

    // ---- Layer 1: cur1(16x32) = x_t(16x57) @ w1^T(57x28, padded) ----
    const float* xw = &xtile[buf][wave * 16 * D_IN];
    v8f acc0 = {}, acc1 = {};
#pragma unroll
    for (int c = 0; c < NCH1; ++c) {
      int k = c * 4 + hi * 2;
      v2f a;
      a.x = (k     < D_IN) ? xw[r * D_IN + k]     : 0.f;
      a.y = (k + 1 < D_IN) ? xw[r * D_IN + k + 1] : 0.f;
      acc0 = wmma_f32_16x16x4(a, bw1[c][0], acc0);
      acc1 = wmma_f32_16x16x4(a, bw1[c][1], acc1);
    }

    // ---- LIF 1 (subtract reset, detached, based on incoming mem) ----
    float* s1w = &s1t[wave][0];
#pragma unroll
    for (int nt = 0; nt < 2; ++nt) {
      int n = nt * 16 + col;
#pragma unroll
      for (int vr = 0; vr < 8; ++vr) {
        float cur = (nt ? acc1[vr] : acc0[vr]) + bias1[nt];
        float m = mem1[nt][vr];
        float reset = (m > THRESH) ? 1.f : 0.f;
        m = BETA * m + cur - reset * THRESH;
        float spk = (m > THRESH) ? 1.f : 0.f;
        mem1[nt][vr] = m;
        if (n < H1) {
          int row = vr + hi * 8;
          s1w[row * H1 + n] = spk;
          outS1[((size_t)t * batch + waveRow + row) * H1 + n] = spk;
        }
      }
    }
    __builtin_amdgcn_wave_barrier();  // order intra-wave LDS store -> load

    // ---- Layer 2: cur2(16x16) = s1(16x28) @ w2^T(28x8, padded) ----
    v8f acc2 = {};
#pragma unroll
    for (int c = 0; c < NCH2; ++c) {
      int k = c * 4 + hi * 2;
      v2f a;
      a.x = (k     < H1) ? s1w[r * H1 + k]     : 0.f;
      a.y = (k + 1 < H1) ? s1w[r * H1 + k + 1] : 0.f;
      acc2 = wmma_f32_16x16x4(a, bw2[c], acc2);
    }

    // ---- LIF 2 ----
    float* s2w = &s2t[wave][0];
#pragma unroll
    for (int vr = 0; vr < 8; ++vr) {
      float cur = acc2[vr] + bias2;
      float m = mem2[vr];
      float reset = (m > THRESH) ? 1.f : 0.f;
      m = BETA * m + cur - reset * THRESH;
      float spk = (m > THRESH) ? 1.f : 0.f;
      mem2[vr] = m;
      if (col < H2) {
        int row = vr + hi * 8;
        s2w[row * 9 + col] = spk;
        outS2[((size_t)t * batch + waveRow + row) * H2 + col] = spk;
      }
    }
    __builtin_amdgcn_wave_barrier();

    // ---- Layer 3: cur3(16x16) = s2(16x8) @ w3^T(8x3, padded) ----
    v8f acc3 = {};
#pragma unroll
    for (int c = 0; c < NCH3; ++c) {
      int k = c * 4 + hi * 2;
      v2f a;
      a.x = (k     < H2) ? s2w[r * 9 + k]     : 0.f;
      a.y = (k + 1 < H2) ? s2w[r * 9 + k + 1] : 0.f;
      acc3 = wmma_f32_16x16x4(a, bw3[c], acc3);
    }

    // ---- LIF 3 + record mem3/spk3 ----
#pragma unroll
    for (int vr = 0; vr < 8; ++vr) {
      float cur = acc3[vr] + bias3;
      float m = mem3[vr];
      float reset = (m > THRESH) ? 1.f : 0.f;
      m = BETA * m + cur - reset * THRESH;
      float spk = (m > THRESH) ? 1.f : 0.f;
      mem3[vr] = m;
      if (col < O_OUT) {
        int row = vr + hi * 8;
        size_t idx = ((size_t)t * batch + waveRow + row) * O_OUT + col;
        outM3[idx] = m;
        outS3[idx] = spk;
      }
    }
  }
}

extern "C" void kernel_launch(void* const* d_in, const int* in_sizes, int n_in,
                              void* d_out, int out_size, void* d_ws, size_t ws_size,
                              hipStream_t stream) {
  const float* x  = (const float*)d_in[0];
  const float* w1 = (const float*)d_in[1];
  const float* b1 = (const float*)d_in[2];
  const float* w2 = (const float*)d_in[3];
  const float* b2 = (const float*)d_in[4];
  const float* w3 = (const float*)d_in[5];
  const float* b3 = (const float*)d_in[6];
  float* out = (float*)d_out;

  const int batch = in_sizes[0] / (T_STEPS * D_IN);   // 65536
  const int grid  = batch / ROWS_PER_BLOCK;           // 1024 blocks of 128 threads

  snn_lif_wmma_kernel<<<grid, THREADS, 0, stream>>>(x, w1, b1, w2, b2, w3, b3, out, batch);
}